// GiG_72988674228247
// MI455X (gfx1250) — compile-verified
//
#include <hip/hip_runtime.h>
#include <cstddef>

// ---------------------------------------------------------------------------
// Types for WMMA f32 16x16x4 (A: 16x4 f32 = 2 VGPRs/lane, C/D: 8 VGPRs/lane)
// ---------------------------------------------------------------------------
typedef __attribute__((ext_vector_type(2))) float v2f;
typedef __attribute__((ext_vector_type(8))) float v8f;

#define CDIV(a, b) (((a) + (b) - 1) / (b))

__device__ __forceinline__ float atomAddF(float* p, float v) {
    return __hip_atomic_fetch_add(p, v, __ATOMIC_RELAXED, __HIP_MEMORY_SCOPE_AGENT);
}

__device__ __forceinline__ float waveRedSum(float v) {
    #pragma unroll
    for (int o = 16; o > 0; o >>= 1) v += __shfl_xor(v, o, 32);
    return v;
}

// ---------------------------------------------------------------------------
// LDS-staged fp32 WMMA GEMM: C[M x NC] = A[M x K] @ B[K x NC] (+ bias).
// NC in {64,128}; K multiple of KCHUNK(=64). One wave computes a 16 x NC strip
// (NC/16 accumulators); 8 waves / 256-thread block -> 128 rows per block.
// B chunk staged in LDS with padded stride so half-wave K-offsets hit
// disjoint bank groups. Branch-free inner loop; EXEC all-1s at every WMMA.
// ---------------------------------------------------------------------------
template <int KCHUNK, int NC>
__global__ __launch_bounds__(256) void gemm_lds_wmma(
    const float* __restrict__ A, const float* __restrict__ Bm,
    const float* __restrict__ bias, float* __restrict__ C, int M, int K) {
    constexpr int NT = NC / 16;      // column tiles per wave
    constexpr int LDB = NC + 8;      // padded LDS stride (2*LDB mod 64 == 16)
    __shared__ float sB[KCHUNK * LDB];

    const int tilesM = (M + 15) >> 4;
    int tm = blockIdx.x * 8 + (threadIdx.x >> 5);
    const bool wactive = (tm < tilesM);
    if (!wactive) tm = tilesM - 1;   // clamp: keep wave alive for LDS fill + barriers
    const int lane = threadIdx.x & 31;
    const int half = lane >> 4;      // 0: K pair {0,1}, 1: K pair {2,3}
    const int l = lane & 15;
    int arow = tm * 16 + l; if (arow >= M) arow = M - 1;
    const float* Abase = A + (size_t)arow * K;

    const v8f vzero = {0.f, 0.f, 0.f, 0.f, 0.f, 0.f, 0.f, 0.f};
    v8f acc[NT];
    #pragma unroll
    for (int t = 0; t < NT; ++t) acc[t] = vzero;

    for (int kc = 0; kc < K; kc += KCHUNK) {
        __syncthreads();
        // cooperative stage of B[kc:kc+KCHUNK, :] into LDS (float4 granularity)
        for (int q = threadIdx.x; q < KCHUNK * (NC / 4); q += 256) {
            int r = q / (NC / 4), c4 = (q % (NC / 4)) * 4;
            *(float4*)&sB[r * LDB + c4] = *(const float4*)&Bm[(size_t)(kc + r) * NC + c4];
        }
        __syncthreads();
        const float* Arow = Abase + kc;
        if (kc + KCHUNK < K) __builtin_prefetch(Arow + KCHUNK, 0, 1);
        #pragma unroll 4
        for (int k0 = 0; k0 < KCHUNK; k0 += 4) {
            const int ka = k0 + 2 * half;
            v2f a;
            a.x = Arow[ka];
            a.y = Arow[ka + 1];
            #pragma unroll
            for (int t = 0; t < NT; ++t) {
                v2f b;
                b.x = sB[ka * LDB + t * 16 + l];
                b.y = sB[(ka + 1) * LDB + t * 16 + l];
                acc[t] = __builtin_amdgcn_wmma_f32_16x16x4_f32(false, a, false, b, (short)0,
                                                               acc[t], false, false);
            }
        }
    }
    if (!wactive) return;
    #pragma unroll
    for (int t = 0; t < NT; ++t) {
        const int bcol = t * 16 + l;
        const float bv = bias ? bias[bcol] : 0.0f;
        #pragma unroll
        for (int r = 0; r < 8; ++r) {
            const int orow = tm * 16 + r + 8 * half;  // C/D layout: VGPR r -> M, M+8
            if (orow < M) C[(size_t)orow * NC + bcol] = acc[t][r] + bv;
        }
    }
}

// ---------------------------------------------------------------------------
// G = Z @ Z^T, Z: [Bn x 64]; branch-free, fully unrolled K=64.
// ---------------------------------------------------------------------------
__global__ __launch_bounds__(256) void gemm_zzt_wmma(const float* __restrict__ Z,
                                                     float* __restrict__ G, int Bn) {
    const int wid = blockIdx.x * 8 + (threadIdx.x >> 5);
    const int tilesN = Bn >> 4;
    const int tm = wid / tilesN, tn = wid % tilesN;
    const int lane = threadIdx.x & 31;
    const int half = lane >> 4, l = lane & 15;
    const float* zr = Z + (size_t)(tm * 16 + l) * 64;   // A fragment rows
    const float* zc = Z + (size_t)(tn * 16 + l) * 64;   // B^T fragment rows
    v8f acc = {0.f, 0.f, 0.f, 0.f, 0.f, 0.f, 0.f, 0.f};
    #pragma unroll
    for (int k0 = 0; k0 < 64; k0 += 4) {
        const int ka = k0 + 2 * half;
        v2f a, b;
        a.x = zr[ka]; a.y = zr[ka + 1];
        b.x = zc[ka]; b.y = zc[ka + 1];
        acc = __builtin_amdgcn_wmma_f32_16x16x4_f32(false, a, false, b, (short)0, acc,
                                                    false, false);
    }
    #pragma unroll
    for (int r = 0; r < 8; ++r) {
        const int orow = tm * 16 + r + 8 * half;
        G[(size_t)orow * Bn + tn * 16 + l] = acc[r];
    }
}

// ---------------------------------------------------------------------------
// Generic guarded WMMA GEMM (used only for the tiny logits GEMM, Nc=8).
// ---------------------------------------------------------------------------
__global__ void gemm_small_wmma(const float* __restrict__ A, const float* __restrict__ Bm,
                                const float* __restrict__ bias, float* __restrict__ C,
                                int M, int K, int Nc) {
    const int wid = blockIdx.x * (blockDim.x >> 5) + (threadIdx.x >> 5);
    const int tilesN = (Nc + 15) >> 4;
    const int tilesM = (M + 15) >> 4;
    if (wid >= tilesM * tilesN) return;  // wave-uniform
    const int tm = wid / tilesN, tn = wid % tilesN;
    const int lane = threadIdx.x & 31;
    const int half = lane >> 4, l = lane & 15;
    int arow = tm * 16 + l; if (arow >= M) arow = M - 1;
    const int bcol = tn * 16 + l;
    const int bc = (bcol < Nc) ? bcol : (Nc - 1);
    const float bmask = (bcol < Nc) ? 1.0f : 0.0f;   // zero pad without branches
    v8f acc = {0.f, 0.f, 0.f, 0.f, 0.f, 0.f, 0.f, 0.f};
    for (int k0 = 0; k0 < K; k0 += 4) {
        const int ka = k0 + 2 * half;
        v2f a, b;
        a.x = A[(size_t)arow * K + ka];
        a.y = A[(size_t)arow * K + ka + 1];
        b.x = Bm[(size_t)ka * Nc + bc] * bmask;
        b.y = Bm[(size_t)(ka + 1) * Nc + bc] * bmask;
        acc = __builtin_amdgcn_wmma_f32_16x16x4_f32(false, a, false, b, (short)0, acc,
                                                    false, false);
    }
    const float bv = (bias != nullptr) ? bias[bc] : 0.0f;
    #pragma unroll
    for (int r = 0; r < 8; ++r) {
        const int orow = tm * 16 + r + 8 * half;
        if (orow < M && bcol < Nc) C[(size_t)orow * Nc + bcol] = acc[r] + bv;
    }
}

// ---------------------------------------------------------------------------
// Small utility / graph kernels
// ---------------------------------------------------------------------------
__global__ void fill_f32(float* __restrict__ p, float v, int n) {
    int i = blockIdx.x * blockDim.x + threadIdx.x;
    if (i < n) p[i] = v;
}

__global__ void deg_count(const int* __restrict__ dst, float* __restrict__ deg, int E) {
    int e = blockIdx.x * blockDim.x + threadIdx.x;
    if (e < E) atomAddF(&deg[dst[e]], 1.0f);
}

__global__ void rsqrt_k(const float* __restrict__ in, float* __restrict__ out, int n) {
    int i = blockIdx.x * blockDim.x + threadIdx.x;
    if (i < n) out[i] = rsqrtf(in[i]);
}

// out[i][f] = hW[i][f] * (1/deg[i]) + bias[f]   (self-loop term; 1/deg = dinv^2)
__global__ void selfloop_init(const float* __restrict__ hw, const float* __restrict__ dinv,
                              const float* __restrict__ bias, float* __restrict__ out, int N) {
    int t = blockIdx.x * blockDim.x + threadIdx.x;
    if (t >= N * 128) return;
    int i = t >> 7, f = t & 127;
    float dv = dinv[i];
    out[t] = hw[t] * dv * dv + bias[f];
}

// scatter: out[dst] += hW[src] * dinv[src]*dinv[dst]; 1 thread = 1 edge x 4 features
__global__ void gcn_scatter(const float* __restrict__ hw, const int* __restrict__ src,
                            const int* __restrict__ dst, const float* __restrict__ dinv,
                            float* __restrict__ out, int E) {
    int t = blockIdx.x * blockDim.x + threadIdx.x;
    int e = t >> 5;
    if (e >= E) return;
    int c = (t & 31) << 2;
    int s = src[e], d = dst[e];
    float w = dinv[s] * dinv[d];
    const float4 h4 = *(const float4*)(hw + (size_t)s * 128 + c);
    float* o = out + (size_t)d * 128 + c;
    atomAddF(o + 0, h4.x * w);
    atomAddF(o + 1, h4.y * w);
    atomAddF(o + 2, h4.z * w);
    atomAddF(o + 3, h4.w * w);
}

// per-row layernorm (+ act: 0 none, 1 relu, 2 leaky 0.2), one wave per row, in-place
__global__ void ln_act(float* __restrict__ x, const float* __restrict__ g,
                       const float* __restrict__ b, int M, int D, int act) {
    int wid = blockIdx.x * (blockDim.x >> 5) + (threadIdx.x >> 5);
    if (wid >= M) return;
    int lane = threadIdx.x & 31;
    float* row = x + (size_t)wid * D;
    float s = 0.f, s2 = 0.f;
    for (int f = lane; f < D; f += 32) { float v = row[f]; s += v; s2 += v * v; }
    s = waveRedSum(s); s2 = waveRedSum(s2);
    float m = s / D;
    float var = fmaxf(s2 / D - m * m, 0.0f);
    float r = rsqrtf(var + 1e-5f);
    for (int f = lane; f < D; f += 32) {
        float v = (row[f] - m) * r * g[f] + b[f];
        if (act == 1) v = v > 0.f ? v : 0.f;
        else if (act == 2) v = v > 0.f ? v : 0.2f * v;
        row[f] = v;
    }
}

__global__ void pool_count(const int* __restrict__ batch, float* __restrict__ counts, int N) {
    int i = blockIdx.x * blockDim.x + threadIdx.x;
    if (i < N) atomAddF(&counts[batch[i]], 1.0f);
}

__global__ void pool_sum(const float* __restrict__ h, const int* __restrict__ batch,
                         float* __restrict__ gemb, int N) {
    int t = blockIdx.x * blockDim.x + threadIdx.x;
    if (t >= N * 128) return;
    int i = t >> 7, f = t & 127;
    atomAddF(&gemb[(size_t)batch[i] * 128 + f], h[t]);
}

__global__ void pool_div(float* __restrict__ gemb, const float* __restrict__ counts, int Bn) {
    int t = blockIdx.x * blockDim.x + threadIdx.x;
    if (t >= Bn * 128) return;
    gemb[t] /= fmaxf(counts[t >> 7], 1.0f);
}

__global__ void rownorm_k(const float* __restrict__ z, float* __restrict__ z2, int Bn, int D) {
    int i = blockIdx.x * blockDim.x + threadIdx.x;
    if (i >= Bn) return;
    float s = 0.f;
    for (int f = 0; f < D; ++f) { float v = z[(size_t)i * D + f]; s += v * v; }
    z2[i] = s;
}

// A = sigmoid(temp * dist + theta + I); dist = -sqrt(max(d2,0)+eps)*(d2!=0)
__global__ void build_A(const float* __restrict__ G, const float* __restrict__ z2,
                        const float* __restrict__ temp, const float* __restrict__ theta,
                        float* __restrict__ Aout, int Bn) {
    int t = blockIdx.x * blockDim.x + threadIdx.x;
    if (t >= Bn * Bn) return;
    int i = t / Bn, j = t % Bn;
    float d2 = fmaxf(z2[i] + z2[j] - 2.0f * G[t], 0.0f);
    if (i == j) d2 = 0.0f;
    float mask = (d2 != 0.0f) ? 1.0f : 0.0f;
    float dist = -sqrtf(d2 + 1.1920929e-07f) * mask;
    float prob = temp[0] * dist + theta[0] + ((i == j) ? 1.0f : 0.0f);
    Aout[t] = 1.0f / (1.0f + expf(-prob));
}

// deg_soft[i] = sum_j A[i][j] * (A[i][j] > 0.5) ; one wave per row
__global__ void degsoft_row(const float* __restrict__ A, float* __restrict__ degsoft, int Bn) {
    int wid = blockIdx.x * (blockDim.x >> 5) + (threadIdx.x >> 5);
    if (wid >= Bn) return;
    int lane = threadIdx.x & 31;
    float s = 0.f;
    for (int j = lane; j < Bn; j += 32) {
        float a = A[(size_t)wid * Bn + j];
        s += (a > 0.5f) ? a : 0.0f;
    }
    s = waveRedSum(s);
    if (lane == 0) degsoft[wid] = s;
}

// wsum[i] = sum_j exp(-(deg[i]-(j+1))^2 / 0.72) ; one wave per row
__global__ void wsum_row(const float* __restrict__ degsoft, float* __restrict__ wsum, int Bn) {
    int wid = blockIdx.x * (blockDim.x >> 5) + (threadIdx.x >> 5);
    if (wid >= Bn) return;
    int lane = threadIdx.x & 31;
    float d = degsoft[wid];
    float s = 0.f;
    for (int j = lane; j < Bn; j += 32) {
        float df = d - (float)(j + 1);
        s += expf(-df * df / 0.72f);
    }
    s = waveRedSum(s);
    if (lane == 0) wsum[wid] = s;
}

// hist[j] = sum_i exp(-(deg[i]-(j+1))^2/0.72)/wsum[i] ; one thread per column
__global__ void hist_k(const float* __restrict__ degsoft, const float* __restrict__ wsum,
                       float* __restrict__ hist, int Bn) {
    int j = blockIdx.x * blockDim.x + threadIdx.x;
    if (j >= Bn) return;
    float c = (float)(j + 1);
    float s = 0.f;
    for (int i = 0; i < Bn; ++i) {
        float df = degsoft[i] - c;
        s += expf(-df * df / 0.72f) / wsum[i];
    }
    hist[j] = s;
}

__global__ void kl_kernel(const float* __restrict__ hist, const float* __restrict__ mu,
                          const float* __restrict__ sigma, float* __restrict__ out, int Bn) {
    __shared__ float red[256];
    const int tid = threadIdx.x;
    auto blockReduce = [&](float v) -> float {
        red[tid] = v; __syncthreads();
        for (int s = 128; s > 0; s >>= 1) { if (tid < s) red[tid] += red[tid + s]; __syncthreads(); }
        float r = red[0]; __syncthreads(); return r;
    };
    float hs = 0.f;
    for (int j = tid; j < Bn; j += 256) hs += hist[j];
    float hsum = blockReduce(hs);
    const float m0 = mu[0], sg = sigma[0];
    const float inv_s2 = 1.0f / (sg * sg);
    float ts = 0.f;
    for (int j = tid; j < Bn; j += 256) { float d = m0 - (float)j; ts += expf(-d * d * inv_s2); }
    float tsum = blockReduce(ts);
    float k = 0.f;
    for (int j = tid; j < Bn; j += 256) {
        float distr = hist[j] / hsum;
        float d = m0 - (float)j;
        float t = expf(-d * d * inv_s2) / tsum;
        k += distr * logf(distr / (t + 1e-8f) + 1e-8f);
    }
    float kl = blockReduce(k);
    if (tid == 0) out[0] = kl;
}

__global__ void aeff_k(const float* __restrict__ A, float* __restrict__ Aeff, int n) {
    int t = blockIdx.x * blockDim.x + threadIdx.x;
    if (t >= n) return;
    float a = A[t];
    Aeff[t] = (a > 0.1f) ? a : 0.0f;
}

// dinvA[i] = rsqrt(max(row_sum(Aeff[i]), 1e-12)) (A_eff symmetric -> colsum==rowsum)
__global__ void rowsum_rsqrt(const float* __restrict__ Aeff, float* __restrict__ dinvA, int Bn) {
    int wid = blockIdx.x * (blockDim.x >> 5) + (threadIdx.x >> 5);
    if (wid >= Bn) return;
    int lane = threadIdx.x & 31;
    float s = 0.f;
    for (int j = lane; j < Bn; j += 32) s += Aeff[(size_t)wid * Bn + j];
    s = waveRedSum(s);
    if (lane == 0) dinvA[wid] = rsqrtf(fmaxf(s, 1e-12f));
}

// column-wise (batch) stats over Bn rows; one wave per column
__global__ void colstats(const float* __restrict__ x, float* __restrict__ colm,
                         float* __restrict__ colv, int Bn, int D) {
    int col = blockIdx.x * (blockDim.x >> 5) + (threadIdx.x >> 5);
    if (col >= D) return;
    int lane = threadIdx.x & 31;
    float s = 0.f, s2 = 0.f;
    for (int r = lane; r < Bn; r += 32) {
        float v = x[(size_t)r * D + col];
        s += v; s2 += v * v;
    }
    s = waveRedSum(s); s2 = waveRedSum(s2);
    if (lane == 0) {
        float m = s / Bn;
        colm[col] = m;
        colv[col] = fmaxf(s2 / Bn - m * m, 0.0f);
    }
}

__global__ void bn_act(float* __restrict__ x, const float* __restrict__ g,
                       const float* __restrict__ b, const float* __restrict__ colm,
                       const float* __restrict__ colv, int Bn, int D, int act) {
    int t = blockIdx.x * blockDim.x + threadIdx.x;
    if (t >= Bn * D) return;
    int f = t % D;
    float v = (x[t] - colm[f]) * rsqrtf(colv[f] + 1e-5f) * g[f] + b[f];
    if (act == 1) v = v > 0.f ? v : 0.f;
    else if (act == 2) v = v > 0.f ? v : 0.2f * v;
    x[t] = v;
}

__global__ void rowscale(const float* __restrict__ in, const float* __restrict__ dinvA,
                         float* __restrict__ out, int n) {
    int t = blockIdx.x * blockDim.x + threadIdx.x;
    if (t >= n) return;
    out[t] = in[t] * dinvA[t >> 7];  // D = 128
}

__global__ void rowscale_bias_act(const float* __restrict__ in, const float* __restrict__ dinvA,
                                  const float* __restrict__ bias, float* __restrict__ out,
                                  int n, int act) {
    int t = blockIdx.x * blockDim.x + threadIdx.x;
    if (t >= n) return;
    int i = t >> 7, f = t & 127;  // D = 128
    float v = dinvA[i] * in[t] + bias[f];
    if (act == 1) v = v > 0.f ? v : 0.f;
    else if (act == 2) v = v > 0.f ? v : 0.2f * v;
    out[t] = v;
}

// ---------------------------------------------------------------------------
// Orchestration
// ---------------------------------------------------------------------------
extern "C" void kernel_launch(void* const* d_in, const int* in_sizes, int n_in,
                              void* d_out, int out_size, void* d_ws, size_t ws_size,
                              hipStream_t stream) {
    (void)in_sizes; (void)n_in; (void)out_size; (void)ws_size;
    const int N = 50000, E = 800000, Bn = 1024;

    const float* x     = (const float*)d_in[0];
    const int*   ei    = (const int*)d_in[1];
    const int*   srcI  = ei;
    const int*   dstI  = ei + E;
    const int*   batch = (const int*)d_in[2];
    // params: nl layers at 3 + 4*i (w,b,lng,lnb); pl at 19 + 4*i; scalars 31..34; cl 35..48
    const float* temp  = (const float*)d_in[31];
    const float* theta = (const float*)d_in[32];
    const float* mu    = (const float*)d_in[33];
    const float* sigma = (const float*)d_in[34];

    float* ws = (float*)d_ws;
    size_t o = 0;
    float* bufA   = ws + o; o += (size_t)N * 128;   // GEMM scratch / later B-level scratch
    float* bufB   = ws + o; o += (size_t)N * 128;   // node features
    float* deg    = ws + o; o += N;
    float* dinv   = ws + o; o += N;
    float* counts = ws + o; o += Bn;
    float* gemb   = ws + o; o += (size_t)Bn * 128;
    float* z2     = ws + o; o += Bn;
    float* degsf  = ws + o; o += Bn;
    float* wsum   = ws + o; o += Bn;
    float* hist   = ws + o; o += Bn;
    float* colm   = ws + o; o += 128;
    float* colv   = ws + o; o += 128;
    float* dinvA  = ws + o; o += Bn;
    // B-level buffers aliased into bufA (free after pooling):
    float* G  = bufA;                           // Bn*Bn (z@z^T, then reused as A_eff)
    float* za = bufA + (size_t)Bn * Bn;         // Bn*128
    float* zb = za + (size_t)Bn * 128;
    float* g0 = zb + (size_t)Bn * 128;
    float* g1 = g0 + (size_t)Bn * 128;
    float* h2 = g1 + (size_t)Bn * 128;

    float* logits = (float*)d_out;
    float* Aout   = logits + (size_t)Bn * 8;
    float* klout  = Aout + (size_t)Bn * Bn;

    auto gemm128 = [&](const float* A, const float* Bm, const float* bias, float* C,
                       int M, int K) {
        gemm_lds_wmma<64, 128><<<CDIV(CDIV(M, 16), 8), 256, 0, stream>>>(A, Bm, bias, C, M, K);
    };
    auto gemm64 = [&](const float* A, const float* Bm, const float* bias, float* C,
                      int M, int K) {
        gemm_lds_wmma<64, 64><<<CDIV(CDIV(M, 16), 8), 256, 0, stream>>>(A, Bm, bias, C, M, K);
    };

    // ---- degrees (shared across all 4 GCN layers) ----
    fill_f32<<<CDIV(N, 256), 256, 0, stream>>>(deg, 1.0f, N);
    deg_count<<<CDIV(E, 256), 256, 0, stream>>>(dstI, deg, E);
    rsqrt_k<<<CDIV(N, 256), 256, 0, stream>>>(deg, dinv, N);

    // ---- F1: 4 GCN layers ----
    const float* hin = x;
    for (int layer = 0; layer < 4; ++layer) {
        const float* W   = (const float*)d_in[3 + 4 * layer];
        const float* bb  = (const float*)d_in[4 + 4 * layer];
        const float* lng = (const float*)d_in[5 + 4 * layer];
        const float* lnb = (const float*)d_in[6 + 4 * layer];
        gemm128(hin, W, nullptr, bufA, N, 128);                            // hW = h @ W
        selfloop_init<<<CDIV(N * 128, 256), 256, 0, stream>>>(bufA, dinv, bb, bufB, N);
        gcn_scatter<<<CDIV(E * 32, 256), 256, 0, stream>>>(bufA, srcI, dstI, dinv, bufB, E);
        ln_act<<<CDIV(N, 8), 256, 0, stream>>>(bufB, lng, lnb, N, 128, layer < 3 ? 1 : 2);
        hin = bufB;
    }

    // ---- mean pool per graph ----
    fill_f32<<<CDIV(Bn, 256), 256, 0, stream>>>(counts, 0.0f, Bn);
    fill_f32<<<CDIV(Bn * 128, 256), 256, 0, stream>>>(gemb, 0.0f, Bn * 128);
    pool_count<<<CDIV(N, 256), 256, 0, stream>>>(batch, counts, N);
    pool_sum<<<CDIV(N * 128, 256), 256, 0, stream>>>(bufB, batch, gemb, N);
    pool_div<<<CDIV(Bn * 128, 256), 256, 0, stream>>>(gemb, counts, Bn);

    // ---- F2: latent MLP ----
    const float* zc = gemb;
    float* zdst[3] = {za, zb, za};
    int kdims[3] = {128, 64, 64};
    for (int i = 0; i < 3; ++i) {
        const float* W   = (const float*)d_in[19 + 4 * i];
        const float* bb  = (const float*)d_in[20 + 4 * i];
        const float* lng = (const float*)d_in[21 + 4 * i];
        const float* lnb = (const float*)d_in[22 + 4 * i];
        gemm64(zc, W, bb, zdst[i], Bn, kdims[i]);
        ln_act<<<CDIV(Bn, 8), 256, 0, stream>>>(zdst[i], lng, lnb, Bn, 64, 2);
        zc = zdst[i];
    }
    rownorm_k<<<CDIV(Bn, 256), 256, 0, stream>>>(zc, z2, Bn, 64);
    gemm_zzt_wmma<<<(64 * 64) / 8, 256, 0, stream>>>(zc, G, Bn);           // z @ z^T
    build_A<<<CDIV(Bn * Bn, 256), 256, 0, stream>>>(G, z2, temp, theta, Aout, Bn);

    // ---- KL term ----
    degsoft_row<<<CDIV(Bn, 8), 256, 0, stream>>>(Aout, degsf, Bn);
    wsum_row<<<CDIV(Bn, 8), 256, 0, stream>>>(degsf, wsum, Bn);
    hist_k<<<CDIV(Bn, 256), 256, 0, stream>>>(degsf, wsum, hist, Bn);
    kl_kernel<<<1, 256, 0, stream>>>(hist, mu, sigma, klout, Bn);

    // ---- F3: classifier over thresholded population graph ----
    float* Aeff = G;
    aeff_k<<<CDIV(Bn * Bn, 256), 256, 0, stream>>>(Aout, Aeff, Bn * Bn);
    rowsum_rsqrt<<<CDIV(Bn, 8), 256, 0, stream>>>(Aeff, dinvA, Bn);

    gemm128(gemb, (const float*)d_in[35], (const float*)d_in[36], g0, Bn, 128);
    colstats<<<CDIV(128, 8), 256, 0, stream>>>(g0, colm, colv, Bn, 128);
    bn_act<<<CDIV(Bn * 128, 256), 256, 0, stream>>>(g0, (const float*)d_in[37],
                                                    (const float*)d_in[38], colm, colv,
                                                    Bn, 128, 2);
    for (int l = 0; l < 2; ++l) {
        const float* W  = (const float*)d_in[39 + 2 * l];
        const float* bb = (const float*)d_in[40 + 2 * l];
        gemm128(g0, W, nullptr, g1, Bn, 128);                              // h = g @ W
        rowscale<<<CDIV(Bn * 128, 256), 256, 0, stream>>>(g1, dinvA, h2, Bn * 128);
        gemm128(Aeff, h2, nullptr, g1, Bn, Bn);                            // A_eff @ (dinv*h)
        rowscale_bias_act<<<CDIV(Bn * 128, 256), 256, 0, stream>>>(g1, dinvA, bb, g0,
                                                                   Bn * 128, 1);
    }
    gemm128(g0, (const float*)d_in[43], (const float*)d_in[44], g1, Bn, 128);
    colstats<<<CDIV(128, 8), 256, 0, stream>>>(g1, colm, colv, Bn, 128);
    bn_act<<<CDIV(Bn * 128, 256), 256, 0, stream>>>(g1, (const float*)d_in[45],
                                                    (const float*)d_in[46], colm, colv,
                                                    Bn, 128, 2);
    gemm_small_wmma<<<CDIV(64, 8), 256, 0, stream>>>(g1, (const float*)d_in[47],
                                                     (const float*)d_in[48], logits,
                                                     Bn, 128, 8);
}